// Encoder_26121991094397
// MI455X (gfx1250) — compile-verified
//
#include <hip/hip_runtime.h>
#include <hip/hip_bf16.h>
#include <math.h>

typedef __attribute__((ext_vector_type(2))) float v2f;
typedef __attribute__((ext_vector_type(8))) float v8f;

#define ISZ 1024          // hidden size (structurally assumed: 256 thr * float4)
#define LSZ 16            // tokens per header unit
#define LN_EPS 1e-12f

__device__ __forceinline__ int find_seg(const int* __restrict__ offs, int B, int t) {
  int lo = 0, hi = B;
  while (hi - lo > 1) {
    int mid = (lo + hi) >> 1;
    if (t >= offs[mid]) lo = mid; else hi = mid;
  }
  return lo;
}

// ---------------- K1: exclusive prefix sum of l_hs -> offs[0..B] ----------------
__global__ void k_offsets(const int* __restrict__ l_hs, int* __restrict__ offs, int B) {
  if (threadIdx.x == 0) {
    int s = 0;
    offs[0] = 0;
    for (int i = 0; i < B; ++i) { s += l_hs[i]; offs[i + 1] = s; }
  }
}

// ---------------- K2: C[M,N] = A[M,K] @ Bm[N,K]^T + bias (fp32 WMMA) ------------
// One 16x16 tile per wave (wave32); block = 4 waves covers 16(M) x 64(N).
__global__ void k_gemm_nt(const float* __restrict__ A, const float* __restrict__ Bm,
                          const float* __restrict__ bias, float* __restrict__ C,
                          int K, int lda, int ldb, int ldc) {
  const int wave = threadIdx.x >> 5;
  const int lane = threadIdx.x & 31;
  const int m0 = blockIdx.y * 16;
  const int n0 = blockIdx.x * 64 + wave * 16;
  const int r  = lane & 15;          // A: row M, B: row N
  const int kh = (lane >> 4) << 1;   // lanes 16-31 carry K+2,K+3

  const float* ap = A  + (size_t)(m0 + r) * lda + kh;
  const float* bp = Bm + (size_t)(n0 + r) * ldb + kh;

  v8f acc = {0.f, 0.f, 0.f, 0.f, 0.f, 0.f, 0.f, 0.f};
#pragma unroll 4
  for (int k = 0; k < K; k += 4) {
    v2f a = *(const v2f*)(ap + k);
    v2f b = *(const v2f*)(bp + k);
    acc = __builtin_amdgcn_wmma_f32_16x16x4_f32(false, a, false, b, (short)0, acc,
                                                false, false);
  }
  const int n  = n0 + r;
  const int mb = m0 + ((lane >> 4) << 3);   // D vgpr i: M = mb + i, N = lane&15
  const float bv = bias[n];
#pragma unroll
  for (int i = 0; i < 8; ++i)
    C[(size_t)(mb + i) * ldc + n] = acc[i] + bv;
}

// ---------------- K3: q[b,k] = sum_d u[b,d] * V_w[d,k]  (NN matvec) -------------
__global__ void k_qproj(const float* __restrict__ u, const float* __restrict__ Vw,
                        float* __restrict__ q) {
  const int b = blockIdx.y;
  const int k = blockIdx.x * 256 + threadIdx.x;   // coalesced over k
  const float* ub = u + (size_t)b * ISZ;
  float acc = 0.f;
#pragma unroll 4
  for (int d = 0; d < ISZ; ++d)
    acc = fmaf(ub[d], Vw[(size_t)d * ISZ + k], acc);
  q[(size_t)b * ISZ + k] = acc;
}

// ---------------- K4: fused scores + softmax + pooling (single wemb pass) -------
// Block = 256 threads, one token t per block; 16x1024 tile held in registers.
__global__ void k_attn_pool(const float* __restrict__ wemb, const float* __restrict__ q,
                            const int* __restrict__ offs, float* __restrict__ pooled,
                            int B) {
  const int t   = blockIdx.x;
  const int tid = threadIdx.x;
  const int b   = find_seg(offs, B, t);
  const int d0  = tid * 4;

  const float4 q4 = *reinterpret_cast<const float4*>(q + (size_t)b * ISZ + d0);
  const float* wb = wemb + ((size_t)t * LSZ) * ISZ + d0;

  float4 w[LSZ];
  float  part[LSZ];
#pragma unroll
  for (int l = 0; l < LSZ; ++l) {
    w[l] = *reinterpret_cast<const float4*>(wb + (size_t)l * ISZ);
    part[l] = w[l].x * q4.x + w[l].y * q4.y + w[l].z * q4.z + w[l].w * q4.w;
  }
  // wave32 butterfly reduce (all 16 scores at once)
#pragma unroll
  for (int off = 16; off >= 1; off >>= 1)
#pragma unroll
    for (int l = 0; l < LSZ; ++l)
      part[l] += __shfl_xor(part[l], off, 32);

  __shared__ float red[8][LSZ];
  __shared__ float fin[LSZ];
  const int wave = tid >> 5, lane = tid & 31;
  if (lane == 0)
#pragma unroll
    for (int l = 0; l < LSZ; ++l) red[wave][l] = part[l];
  __syncthreads();
  if (tid < LSZ) {
    float s = 0.f;
#pragma unroll
    for (int wv = 0; wv < 8; ++wv) s += red[wv][tid];
    fin[tid] = s;     // note: +V_b·u term dropped — softmax is shift-invariant
  }
  __syncthreads();

  // every thread redundantly computes the 16-way softmax (cheap)
  float sc[LSZ];
  float m = -3.402823466e38f;
#pragma unroll
  for (int l = 0; l < LSZ; ++l) { sc[l] = fin[l]; m = fmaxf(m, sc[l]); }
  float ssum = 0.f;
#pragma unroll
  for (int l = 0; l < LSZ; ++l) { sc[l] = __expf(sc[l] - m); ssum += sc[l]; }
  const float inv = 1.0f / ssum;

  float4 p = {0.f, 0.f, 0.f, 0.f};
#pragma unroll
  for (int l = 0; l < LSZ; ++l) {
    const float a = sc[l] * inv;
    p.x = fmaf(a, w[l].x, p.x);
    p.y = fmaf(a, w[l].y, p.y);
    p.z = fmaf(a, w[l].z, p.z);
    p.w = fmaf(a, w[l].w, p.w);
  }
  *reinterpret_cast<float4*>(pooled + (size_t)t * ISZ + d0) = p;
}

// ---------------- K5: h = [pooled | ctx[seg]] @ enc_w^T + enc_b (fp32 WMMA) -----
__global__ void k_enc_gemm(const float* __restrict__ pooled, const float* __restrict__ ctx,
                           const float* __restrict__ encw, const float* __restrict__ encb,
                           const int* __restrict__ offs, float* __restrict__ h, int B) {
  const int wave = threadIdx.x >> 5;
  const int lane = threadIdx.x & 31;
  const int t0 = blockIdx.y * 16;
  const int n0 = blockIdx.x * 64 + wave * 16;
  const int r  = lane & 15;
  const int kh = (lane >> 4) << 1;

  const int t = t0 + r;
  const int b = find_seg(offs, B, t);
  const float* ap1 = pooled + (size_t)t * ISZ + kh;            // K in [0,1024)
  const float* ap2 = ctx    + (size_t)b * ISZ + kh;            // K in [1024,2048)
  const float* bp  = encw   + (size_t)(n0 + r) * (2 * ISZ) + kh;

  v8f acc = {0.f, 0.f, 0.f, 0.f, 0.f, 0.f, 0.f, 0.f};
#pragma unroll 4
  for (int k = 0; k < ISZ; k += 4) {
    v2f a  = *(const v2f*)(ap1 + k);
    v2f bb = *(const v2f*)(bp + k);
    acc = __builtin_amdgcn_wmma_f32_16x16x4_f32(false, a, false, bb, (short)0, acc,
                                                false, false);
  }
#pragma unroll 4
  for (int k = 0; k < ISZ; k += 4) {
    v2f a  = *(const v2f*)(ap2 + k);
    v2f bb = *(const v2f*)(bp + ISZ + k);
    acc = __builtin_amdgcn_wmma_f32_16x16x4_f32(false, a, false, bb, (short)0, acc,
                                                false, false);
  }
  const int n  = n0 + r;
  const int mb = t0 + ((lane >> 4) << 3);
  const float bv = encb[n];
#pragma unroll
  for (int i = 0; i < 8; ++i)
    h[(size_t)(mb + i) * ISZ + n] = acc[i] + bv;
}

// ---------------- K6: ragged scatter + TF-style LayerNorm -----------------------
__global__ void k_ln_scatter(const float* __restrict__ h, const int* __restrict__ offs,
                             const int* __restrict__ l_hs, const float* __restrict__ gamma,
                             const float* __restrict__ beta, float* __restrict__ out,
                             int maxL) {
  const int b   = blockIdx.y;
  const int p   = blockIdx.x;
  const int tid = threadIdx.x;
  const int d0  = tid * 4;
  const size_t obase = ((size_t)b * maxL + p) * ISZ;

  const int len = l_hs[b];               // block-uniform branch
  if (p >= len) {                         // zero row: LN(0) = 0/sqrt(eps) -> beta
    const float4 be = *reinterpret_cast<const float4*>(beta + d0);
    *reinterpret_cast<float4*>(out + obase + d0) = be;
    return;
  }
  const int t = offs[b] + p;
  const float4 x = *reinterpret_cast<const float4*>(h + (size_t)t * ISZ + d0);
  float s  = x.x + x.y + x.z + x.w;
  float s2 = x.x * x.x + x.y * x.y + x.z * x.z + x.w * x.w;
#pragma unroll
  for (int off = 16; off >= 1; off >>= 1) {
    s  += __shfl_xor(s, off, 32);
    s2 += __shfl_xor(s2, off, 32);
  }
  __shared__ float rs[8], rs2[8];
  __shared__ float mu_s, rstd_s;
  const int wave = tid >> 5, lane = tid & 31;
  if (lane == 0) { rs[wave] = s; rs2[wave] = s2; }
  __syncthreads();
  if (tid == 0) {
    float ts = 0.f, ts2 = 0.f;
#pragma unroll
    for (int wv = 0; wv < 8; ++wv) { ts += rs[wv]; ts2 += rs2[wv]; }
    const float mu  = ts / (float)ISZ;
    const float var = ts2 / (float)ISZ - mu * mu;
    mu_s = mu;
    rstd_s = rsqrtf(var + LN_EPS);
  }
  __syncthreads();
  const float mu = mu_s, rstd = rstd_s;
  const float4 g  = *reinterpret_cast<const float4*>(gamma + d0);
  const float4 be = *reinterpret_cast<const float4*>(beta + d0);
  float4 o;
  o.x = fmaf(g.x * (x.x - mu), rstd, be.x);
  o.y = fmaf(g.y * (x.y - mu), rstd, be.y);
  o.z = fmaf(g.z * (x.z - mu), rstd, be.z);
  o.w = fmaf(g.w * (x.w - mu), rstd, be.w);
  *reinterpret_cast<float4*>(out + obase + d0) = o;
}

extern "C" void kernel_launch(void* const* d_in, const int* in_sizes, int n_in,
                              void* d_out, int out_size, void* d_ws, size_t ws_size,
                              hipStream_t stream) {
  const float* ctx   = (const float*)d_in[0];
  const float* wemb  = (const float*)d_in[1];
  // d_in[2] = l_hpu (unused by forward), d_in[4] = max_l device scalar (derived on host)
  const int*   l_hs  = (const int*)d_in[3];
  const float* U_w   = (const float*)d_in[5];
  const float* U_b   = (const float*)d_in[6];
  const float* V_w   = (const float*)d_in[7];
  // d_in[8] = V_b: provably irrelevant (adds a constant across the softmax axis)
  const float* enc_w = (const float*)d_in[9];
  const float* enc_b = (const float*)d_in[10];
  const float* gamma = (const float*)d_in[11];
  const float* beta  = (const float*)d_in[12];
  float* out = (float*)d_out;

  const int IS   = in_sizes[6];               // 1024
  const int Bn   = in_sizes[3];               // 64
  const int Tn   = in_sizes[2];               // 4064 (multiple of 16)
  const int maxL = out_size / (Bn * IS);      // 95

  // workspace layout
  char*  ws     = (char*)d_ws;
  int*   offs   = (int*)ws;                                            // B+1 ints
  float* u      = (float*)(ws + 1024);                                 // B*IS
  float* q      = (float*)(ws + 1024 + (size_t)Bn * IS * 4);           // B*IS
  float* pooled = (float*)(ws + 1024 + 2 * (size_t)Bn * IS * 4);       // T*IS
  float* h      = (float*)(ws + 1024 + 2 * (size_t)Bn * IS * 4
                               + (size_t)Tn * IS * 4);                 // T*IS
  (void)ws_size; (void)n_in;

  // 1) ragged offsets
  k_offsets<<<1, 32, 0, stream>>>(l_hs, offs, Bn);
  // 2) u = ctx @ U_w^T + U_b   (M=B, N=IS, K=IS)  [WMMA f32]
  k_gemm_nt<<<dim3(IS / 64, Bn / 16), 128, 0, stream>>>(ctx, U_w, U_b, u, IS, IS, IS, IS);
  // 3) q = u @ V_w             (V_b dropped: softmax shift-invariant)
  k_qproj<<<dim3(IS / 256, Bn), 256, 0, stream>>>(u, V_w, q);
  // 4) fused scores/softmax/pool: single streaming pass over wemb_hpu (266 MB)
  k_attn_pool<<<Tn, 256, 0, stream>>>(wemb, q, offs, pooled, Bn);
  // 5) h = [pooled | ctx[seg]] @ enc_w^T + enc_b  (M=T, N=IS, K=2*IS) [WMMA f32]
  k_enc_gemm<<<dim3(IS / 64, Tn / 16), 128, 0, stream>>>(pooled, ctx, enc_w, enc_b,
                                                         offs, h, Bn);
  // 6) LayerNorm + scatter into padded [B, maxL, IS]
  k_ln_scatter<<<dim3(maxL, Bn), 256, 0, stream>>>(h, offs, l_hs, gamma, beta, out, maxL);
}